// QLSTM_65481071404161
// MI455X (gfx1250) — compile-verified
//
#include <hip/hip_runtime.h>
#include <math.h>

typedef __attribute__((ext_vector_type(2))) float v2f;
typedef __attribute__((ext_vector_type(8))) float v8f;

#define TT 128
#define BB 2048
#define DD 128
#define ROWS (TT * BB)          // 262144 rows of the fused GEMM
#define NTILES (ROWS / 16)      // 16384 16-row tiles

// ---------------------------------------------------------------------------
// Phase 1: Z[r][0:16] = X[r][0:128] @ Wx^T + bias   (fp32 WMMA 16x16x4)
//   cols 0..3 -> Wf rows 0..3, 4..7 -> Wi, 8..11 -> Wu, 12..15 -> Wo
//   (only the first 128 of the 132 weight columns: the x part)
// One wave computes one 16x16 output tile, accumulating K=128 in 32 WMMAs.
// ---------------------------------------------------------------------------
__global__ __launch_bounds__(256) void qlstm_gemm_wmma(
    const float* __restrict__ X,
    const float* __restrict__ Wf, const float* __restrict__ Wi,
    const float* __restrict__ Wu, const float* __restrict__ Wo,
    const float* __restrict__ bf, const float* __restrict__ bi,
    const float* __restrict__ bu, const float* __restrict__ bo,
    float* __restrict__ Z)
{
    const int lane = threadIdx.x & 31;
    const int wave = threadIdx.x >> 5;
    const int tile = blockIdx.x * 8 + wave;
    if (tile >= NTILES) return;
    const long row0 = (long)tile * 16;

    const int n    = lane & 15;   // output column (B-matrix N / C column)
    const int hgrp = lane >> 4;   // 0: K pair {0,1}; 1: K pair {2,3}
    const int gate = n >> 2;
    const int r    = n & 3;

    const float* Wsel = (gate == 0) ? Wf : (gate == 1) ? Wi : (gate == 2) ? Wu : Wo;
    const float* bsel = (gate == 0) ? bf : (gate == 1) ? bi : (gate == 2) ? bu : bo;
    const float* wrow = Wsel + r * 132;          // row n of W, first 128 cols = Wx
    const float  bias = bsel[r];

    // C init = bias (bias depends only on column n, constant per lane)
    v8f acc;
#pragma unroll
    for (int v = 0; v < 8; ++v) acc[v] = bias;

    // A fragment row for this lane (lanes l and l+16 share row m = l&15)
    const float* arow = X + (row0 + n) * DD;

#pragma unroll 8
    for (int kk = 0; kk < 32; ++kk) {
        const int k0 = kk * 4 + hgrp * 2;
        // A: 16x4 tile, lane l -> A[m][k0], A[m][k0+1]   (8B aligned)
        v2f afrag = *(const v2f*)(arow + k0);
        // B: 4x16 tile, lane l -> B[k0][n], B[k0+1][n] = W[n][k0], W[n][k0+1]
        v2f bfrag = *(const v2f*)(wrow + k0);
        acc = __builtin_amdgcn_wmma_f32_16x16x4_f32(
            /*neg_a=*/false, afrag, /*neg_b=*/false, bfrag,
            /*c_mod=*/(short)0, acc, /*reuse_a=*/false, /*reuse_b=*/false);
    }

    // C/D layout: VGPR v holds C[row0 + v + 8*hgrp][n]
    float* zout = Z + row0 * 16;
#pragma unroll
    for (int v = 0; v < 8; ++v) {
        const int m = v + hgrp * 8;
        zout[(long)m * 16 + n] = acc[v];
    }
}

// ---------------------------------------------------------------------------
// Phase 2: sequential scan over T. One thread per batch element (2048 chains).
// Quantum layer collapsed analytically:
//   x0=cos(a0+th0), x1=cos(a1)*cos(th1), x2=cos(th2), x3=cos(a3+th3)
//   qlayer = [x1*x2*x3, x0*x1, x0*x1*x2, x0*x1*x2*x3]   (a2 is a global phase)
// ---------------------------------------------------------------------------
__global__ __launch_bounds__(128) void qlstm_scan(
    const float* __restrict__ Z,
    const float* __restrict__ Wf, const float* __restrict__ Wi,
    const float* __restrict__ Wu, const float* __restrict__ Wo,
    const float* __restrict__ theta_f, const float* __restrict__ theta_i,
    const float* __restrict__ theta_u, const float* __restrict__ theta_o,
    float* __restrict__ out)
{
    const int b = blockIdx.x * blockDim.x + threadIdx.x;
    if (b >= BB) return;

    // Recurrent weights Wh[n][j] = W_gate[(n&3)*132 + 128 + j]  (uniform -> SMEM)
    const float* Ws[4] = {Wf, Wi, Wu, Wo};
    float wh[16][4];
#pragma unroll
    for (int g = 0; g < 4; ++g)
#pragma unroll
        for (int r = 0; r < 4; ++r)
#pragma unroll
            for (int j = 0; j < 4; ++j)
                wh[g * 4 + r][j] = Ws[g][r * 132 + 128 + j];

    const float* Th[4] = {theta_f, theta_i, theta_u, theta_o};
    float th0[4], th3[4], ct1[4], ct2[4];
#pragma unroll
    for (int g = 0; g < 4; ++g) {
        th0[g] = Th[g][0];
        ct1[g] = __cosf(Th[g][1]);
        ct2[g] = __cosf(Th[g][2]);
        th3[g] = Th[g][3];
    }

    float hv[4] = {0.f, 0.f, 0.f, 0.f};
    float cv[4] = {0.f, 0.f, 0.f, 0.f};

    for (int t = 0; t < TT; ++t) {
        const float* zp = Z + ((long)t * BB + b) * 16;
        float ang[16];
#pragma unroll
        for (int n = 0; n < 16; ++n) {
            float a = zp[n];
#pragma unroll
            for (int j = 0; j < 4; ++j) a = fmaf(hv[j], wh[n][j], a);
            ang[n] = a;
        }

        float gv[4][4];
#pragma unroll
        for (int g = 0; g < 4; ++g) {
            const float x0 = __cosf(ang[g * 4 + 0] + th0[g]);
            const float x1 = __cosf(ang[g * 4 + 1]) * ct1[g];
            const float x2 = ct2[g];
            const float x3 = __cosf(ang[g * 4 + 3] + th3[g]);
            gv[g][0] = x1 * x2 * x3;
            gv[g][1] = x0 * x1;
            gv[g][2] = x0 * x1 * x2;
            gv[g][3] = x0 * x1 * x2 * x3;
        }

#pragma unroll
        for (int w = 0; w < 4; ++w) {
            const float F = 1.0f / (1.0f + __expf(-gv[0][w]));
            const float I = 1.0f / (1.0f + __expf(-gv[1][w]));
            const float U = 1.0f - 2.0f / (__expf(2.0f * gv[2][w]) + 1.0f); // tanh
            const float O = 1.0f / (1.0f + __expf(-gv[3][w]));
            cv[w] = F * cv[w] + I * U;
            const float tc = 1.0f - 2.0f / (__expf(2.0f * cv[w]) + 1.0f);  // tanh(c)
            hv[w] = O * tc;
        }

        float4 hs; hs.x = hv[0]; hs.y = hv[1]; hs.z = hv[2]; hs.w = hv[3];
        *(float4*)(out + ((long)t * BB + b) * 4) = hs;   // stacked[t,b,:]
    }

    // hx, cx appended after stacked
    float4 hs; hs.x = hv[0]; hs.y = hv[1]; hs.z = hv[2]; hs.w = hv[3];
    float4 cs; cs.x = cv[0]; cs.y = cv[1]; cs.z = cv[2]; cs.w = cv[3];
    *(float4*)(out + (long)TT * BB * 4 + (long)b * 4) = hs;
    *(float4*)(out + (long)TT * BB * 4 + (long)BB * 4 + (long)b * 4) = cs;
}

// ---------------------------------------------------------------------------
extern "C" void kernel_launch(void* const* d_in, const int* in_sizes, int n_in,
                              void* d_out, int out_size, void* d_ws, size_t ws_size,
                              hipStream_t stream)
{
    const float* X       = (const float*)d_in[0];
    const float* Wf      = (const float*)d_in[1];
    const float* bf      = (const float*)d_in[2];
    const float* Wi      = (const float*)d_in[3];
    const float* bi      = (const float*)d_in[4];
    const float* Wu      = (const float*)d_in[5];
    const float* bu      = (const float*)d_in[6];
    const float* Wo      = (const float*)d_in[7];
    const float* bo      = (const float*)d_in[8];
    const float* theta_f = (const float*)d_in[9];
    const float* theta_i = (const float*)d_in[10];
    const float* theta_u = (const float*)d_in[11];
    const float* theta_o = (const float*)d_in[12];

    float* Z   = (float*)d_ws;        // 262144 * 16 * 4B = 16 MB scratch
    float* out = (float*)d_out;

    // Phase 1: 16384 tiles, 8 waves (tiles) per 256-thread block
    qlstm_gemm_wmma<<<NTILES / 8, 256, 0, stream>>>(
        X, Wf, Wi, Wu, Wo, bf, bi, bu, bo, Z);

    // Phase 2: one thread per batch chain
    qlstm_scan<<<BB / 128, 128, 0, stream>>>(
        Z, Wf, Wi, Wu, Wo, theta_f, theta_i, theta_u, theta_o, out);
}